// RNNCell_61581241090198
// MI455X (gfx1250) — compile-verified
//
#include <hip/hip_runtime.h>
#include <hip/hip_bf16.h>

// RNN: h_t = tanh(W_x x_t + W_h h_{t-1} + b), B=256, T=512, I=H=256, fp32.
// Phase 1: Xproj = x @ W_x^T  written in-place into d_out (same shape [B,T,H]).
// Phase 2: 16 workgroups (one per 16-row batch tile) iterate t=0..511,
//          W_h held in VGPRs, h double-buffered in LDS, fp32 WMMA 16x16x4,
//          next-step C prefetch + split workgroup barrier to hide latency.

typedef __attribute__((ext_vector_type(2))) float v2f;
typedef __attribute__((ext_vector_type(4))) float v4f;
typedef __attribute__((ext_vector_type(8))) float v8f;

#define B_ 256
#define T_ 512
#define I_ 256
#define H_ 256
#define LDSROW 260  // padded fp32 row stride: (4m+k)%64 distinct -> no bank conflicts

// ---------------------------------------------------------------------------
// Phase 1: Xproj[M,N] = sum_k x[M,k] * Wx[N,k]   (M = B*T flat, N = H, K = I)
// One workgroup = 16-row M tile staged in LDS; 16 waves each own one N tile.
// ---------------------------------------------------------------------------
__global__ __launch_bounds__(512) void rnn_xproj_kernel(
    const float* __restrict__ x, const float* __restrict__ Wx,
    float* __restrict__ out) {
  __shared__ float atile[16 * LDSROW];
  const int tid  = threadIdx.x;
  const int lane = tid & 31;
  const int wave = tid >> 5;                 // 0..15 -> N tile index
  const long M0  = (long)blockIdx.x * 16;    // flat row base in B*T

  // Stage 16x256 fp32 A tile into LDS (coalesced: 8 contiguous floats/thread).
  {
    const int idx = tid * 8;                 // 4096 floats / 512 threads
    const int r   = idx >> 8;
    const int i   = idx & 255;
    const float* src = x + (M0 + r) * (long)I_ + i;
    *(v4f*)(&atile[r * LDSROW + i])     = *(const v4f*)(src);
    *(v4f*)(&atile[r * LDSROW + i + 4]) = *(const v4f*)(src + 4);
  }
  __syncthreads();

  // B fragment: B(k,n) = Wx[n*I + k]; lane -> n, lane/16 selects K pair.
  const int n     = (wave << 4) + (lane & 15);
  const int khalf = (lane >> 4) << 1;        // 0 or 2
  v2f bfrag[64];
#pragma unroll
  for (int kk = 0; kk < 64; ++kk) {
    const int kb = kk * 4 + khalf;
    bfrag[kk].x = Wx[n * I_ + kb + 0];
    bfrag[kk].y = Wx[n * I_ + kb + 1];
  }

  const int m = lane & 15;                   // A row
  v8f acc = {};
#pragma unroll
  for (int kk = 0; kk < 64; ++kk) {
    const int kb = kk * 4 + khalf;
    v2f a;
    a.x = atile[m * LDSROW + kb + 0];
    a.y = atile[m * LDSROW + kb + 1];
    acc = __builtin_amdgcn_wmma_f32_16x16x4_f32(false, a, false, bfrag[kk],
                                                (short)0, acc, false, false);
  }

  // C/D layout: lane%16 = N, VGPR v -> M = v + 8*(lane/16).
  const int mb = (lane >> 4) << 3;
#pragma unroll
  for (int v = 0; v < 8; ++v)
    out[(M0 + mb + v) * (long)H_ + n] = acc[v];
}

// ---------------------------------------------------------------------------
// Phase 2: recurrence. grid = B/16 workgroups; each WG owns batch rows
// [b0, b0+16) and iterates all T steps. W_h in registers, h in LDS (2 bufs).
// Per step: prefetch next C tile first, 64 chained WMMAs, tanh, LDS write,
// split barrier (signal -> global store -> wait) to overlap stores.
// ---------------------------------------------------------------------------
__global__ __launch_bounds__(512) void rnn_recur_kernel(
    const float* __restrict__ Wh, const float* __restrict__ bias,
    float* __restrict__ out) {
  __shared__ float hbuf[2][16 * LDSROW];
  const int tid  = threadIdx.x;
  const int lane = tid & 31;
  const int wave = tid >> 5;
  const int b0   = blockIdx.x * 16;

  const int n     = (wave << 4) + (lane & 15);
  const int khalf = (lane >> 4) << 1;

  // Preload W_h B fragments once: B(k,n) = Wh[n*H + k]. 128 VGPRs, reused 512x.
  v2f bfrag[64];
#pragma unroll
  for (int kk = 0; kk < 64; ++kk) {
    const int kb = kk * 4 + khalf;
    bfrag[kk].x = Wh[n * H_ + kb + 0];
    bfrag[kk].y = Wh[n * H_ + kb + 1];
  }
  const float bn = bias[n];

  // h_0 = 0
  {
    const int idx = tid * 8;
    const int r   = idx >> 8;
    const int i   = idx & 255;
#pragma unroll
    for (int j = 0; j < 8; ++j) hbuf[0][r * LDSROW + i + j] = 0.0f;
  }
  __syncthreads();

  const int m  = lane & 15;
  const int mb = (lane >> 4) << 3;

  // Row base pointers for this thread's 8 C/D elements (column n fixed).
  // out index: ((b0+mb+v)*T + t)*H + n
  long rowbase[8];
#pragma unroll
  for (int v = 0; v < 8; ++v)
    rowbase[v] = ((long)(b0 + mb + v) * T_) * H_ + n;

  // Preload C for t = 0.
  v8f cin;
#pragma unroll
  for (int v = 0; v < 8; ++v) cin[v] = out[rowbase[v]];

  int p = 0;
  for (int t = 0; t < T_; ++t) {
    v8f acc;
#pragma unroll
    for (int v = 0; v < 8; ++v) acc[v] = cin[v] + bn;

    // Prefetch next step's Xproj tile before the WMMA chain (clamped at end;
    // the clamped redundant read of slot t=T-1 is discarded).
    const int tn = (t + 1 < T_) ? (t + 1) : (T_ - 1);
    v8f cnext;
#pragma unroll
    for (int v = 0; v < 8; ++v) cnext[v] = out[rowbase[v] + (long)tn * H_];

    // acc += h_{t-1} @ W_h^T for this 16x16 tile, K = 256 in steps of 4.
#pragma unroll
    for (int kk = 0; kk < 64; ++kk) {
      const int kb = kk * 4 + khalf;
      v2f a;
      a.x = hbuf[p][m * LDSROW + kb + 0];
      a.y = hbuf[p][m * LDSROW + kb + 1];
      acc = __builtin_amdgcn_wmma_f32_16x16x4_f32(false, a, false, bfrag[kk],
                                                  (short)0, acc, false, false);
    }

#pragma unroll
    for (int v = 0; v < 8; ++v) acc[v] = tanhf(acc[v]);

    // h_t into the other LDS buffer (needed by all waves next step).
#pragma unroll
    for (int v = 0; v < 8; ++v)
      hbuf[1 - p][(mb + v) * LDSROW + n] = acc[v];

    // Split barrier: make LDS writes visible, signal arrival, then do the
    // global h_t stores while other waves finish, then wait.
    asm volatile("s_wait_dscnt 0x0" ::: "memory");
    asm volatile("s_barrier_signal -1" ::: "memory");
#pragma unroll
    for (int v = 0; v < 8; ++v)
      out[rowbase[v] + (long)t * H_] = acc[v];
    asm volatile("s_barrier_wait -1" ::: "memory");

    cin = cnext;
    p ^= 1;
  }
}

// ---------------------------------------------------------------------------
extern "C" void kernel_launch(void* const* d_in, const int* in_sizes, int n_in,
                              void* d_out, int out_size, void* d_ws,
                              size_t ws_size, hipStream_t stream) {
  const float* x  = (const float*)d_in[0];  // [B,T,I]
  const float* Wx = (const float*)d_in[1];  // [H,I] (einsum 'hi')
  const float* Wh = (const float*)d_in[2];  // [H,H] (einsum 'hk')
  const float* b  = (const float*)d_in[3];  // [H]
  float* out = (float*)d_out;               // [B,T,H]

  // Phase 1: (B*T)/16 = 8192 row tiles, 512 threads (16 waves) each.
  rnn_xproj_kernel<<<(B_ * T_) / 16, 512, 0, stream>>>(x, Wx, out);
  // Phase 2: B/16 = 16 workgroups carry the recurrence.
  rnn_recur_kernel<<<B_ / 16, 512, 0, stream>>>(Wh, b, out);
}